// NCEAverage_54838142435841
// MI455X (gfx1250) — compile-verified
//
#include <hip/hip_runtime.h>

typedef float v2f __attribute__((ext_vector_type(2)));
typedef float v4f __attribute__((ext_vector_type(4)));
typedef float v8f __attribute__((ext_vector_type(8)));

#define B_    64
#define K_    16384           // last valid k index (K+1 = 16385 slots)
#define KP1   16385
#define D_    128
#define N_    500000
#define INV_T (1.0f / 0.07f)
#define MOM_  0.5f

// ---------------------------------------------------------------------------
// Kernel 1: gathered scoring via V_WMMA_F32_16X16X4_F32.
// One wave handles one batch row b and a tile of 16 k-slots, both banks.
//   out_ab[b,k] = dot(mem_l [idx[b,k]], ab[b]) / T
//   out_l [b,k] = dot(mem_ab[idx[b,k]], l [b]) / T
// A-matrix (16x4 f32, ISA layout): lanes 0-15 -> K=0,1 ; lanes 16-31 -> K=2,3.
// B-operand: the feature chunk is broadcast to ALL lanes -> every column of C
// holds the same dot product (columns are independent), so no masking VALU is
// needed and the result is insensitive to the exact N<->lane mapping.
// C/D (16x16 f32): lanes 0-15 VGPR v -> M=v, lanes 16-31 -> M=v+8; read lane 0
// (M=0..7) and lane 16 (M=8..15).
// ---------------------------------------------------------------------------
__global__ __launch_bounds__(256) void nce_score_kernel(
    const float* __restrict__ l, const float* __restrict__ ab,
    const float* __restrict__ mem_l, const float* __restrict__ mem_ab,
    const int* __restrict__ idx,
    float* __restrict__ out_l, float* __restrict__ out_ab)
{
    __shared__ float s_l[D_];
    __shared__ float s_ab[D_];

    const int b   = blockIdx.y;
    const int tid = threadIdx.x;

    if (tid < D_)            s_l[tid]        = l[b * D_ + tid];
    else if (tid < 2 * D_)   s_ab[tid - D_]  = ab[b * D_ + (tid - D_)];
    __syncthreads();

    const int wave = tid >> 5;
    const int lane = tid & 31;
    const int tile = blockIdx.x * 8 + wave;        // 16-k tile
    const int k0   = tile * 16;
    if (k0 >= KP1) return;                          // wave-uniform exit

    const int m = lane & 15;                        // row within tile
    int kk = k0 + m; if (kk > K_) kk = K_;          // clamp ragged last tile
    const int row = idx[b * KP1 + kk];              // gather index

    const int  sel  = (lane >> 4) << 1;             // 0 (K=0,1) or 2 (K=2,3)
    const long rb   = (long)row * D_ + sel;
    const float* __restrict__ pl  = mem_l  + rb;
    const float* __restrict__ pab = mem_ab + rb;

    v8f acc_l  = {};   // mem_ab x l  -> out_l
    v8f acc_ab = {};   // mem_l  x ab -> out_ab

    #pragma unroll 4
    for (int d = 0; d < D_; d += 4) {
        v2f aL = *(const v2f*)(pab + d);            // A from memory_ab
        v2f aA = *(const v2f*)(pl  + d);            // A from memory_l
        v2f bL = *(const v2f*)(s_l  + d + sel);     // feature chunk, broadcast
        v2f bA = *(const v2f*)(s_ab + d + sel);
        acc_l  = __builtin_amdgcn_wmma_f32_16x16x4_f32(
                     false, aL, false, bL, (short)0, acc_l,  false, false);
        acc_ab = __builtin_amdgcn_wmma_f32_16x16x4_f32(
                     false, aA, false, bA, (short)0, acc_ab, false, false);
    }

    // Every column of C holds the scores; lane 0 -> M=0..7, lane 16 -> M=8..15.
    if (m == 0) {
        const int mbase = k0 + ((lane >> 4) << 3);
        #pragma unroll
        for (int i = 0; i < 8; ++i) {
            const int kidx = mbase + i;
            if (kidx <= K_) {
                out_l [b * KP1 + kidx] = acc_l[i]  * INV_T;
                out_ab[b * KP1 + kidx] = acc_ab[i] * INV_T;
            }
        }
    }
}

// ---------------------------------------------------------------------------
// Kernel 2: stream both memory banks to the output (new_memory_* baseline).
// Pure streaming copy -> non-temporal hints so we don't thrash L2 (which is
// busy serving the 256MB-working-set gather of kernel 1).
// ---------------------------------------------------------------------------
__global__ __launch_bounds__(256) void bank_copy_kernel(
    const v4f* __restrict__ src_l, const v4f* __restrict__ src_ab,
    v4f* __restrict__ dst_l, v4f* __restrict__ dst_ab)
{
    const long n4 = (long)N_ * (D_ / 4);
    const long stride = (long)gridDim.x * blockDim.x;
    for (long i = (long)blockIdx.x * blockDim.x + threadIdx.x; i < n4; i += stride) {
        v4f a = __builtin_nontemporal_load(src_l  + i);
        v4f c = __builtin_nontemporal_load(src_ab + i);
        __builtin_nontemporal_store(a, dst_l  + i);
        __builtin_nontemporal_store(c, dst_ab + i);
    }
}

// ---------------------------------------------------------------------------
// Kernel 3: EMA + L2-normalize + scatter of the B updated rows.
// One wave32 per batch row; lane holds 4 floats (32*4 = D); shfl reduction.
// Runs after the copy (stream-ordered) so it overwrites rows y.
// ---------------------------------------------------------------------------
__global__ __launch_bounds__(64) void ema_update_kernel(
    const float* __restrict__ l, const float* __restrict__ ab,
    const float* __restrict__ mem_l, const float* __restrict__ mem_ab,
    const int* __restrict__ y,
    float* __restrict__ new_l, float* __restrict__ new_ab)
{
    const int gtid = blockIdx.x * blockDim.x + threadIdx.x;
    const int w    = gtid >> 5;            // batch row
    const int lane = threadIdx.x & 31;
    if (w >= B_) return;

    const int  row = y[w];
    const int  c   = lane * 4;
    const long mo  = (long)row * D_ + c;
    const int  fo  = w * D_ + c;

    v4f pl = *(const v4f*)(mem_l  + mo);
    v4f pa = *(const v4f*)(mem_ab + mo);
    v4f fl = *(const v4f*)(l  + fo);
    v4f fa = *(const v4f*)(ab + fo);

    v4f ql = pl * MOM_ + fl * (1.0f - MOM_);
    v4f qa = pa * MOM_ + fa * (1.0f - MOM_);

    float ssl = ql.x * ql.x + ql.y * ql.y + ql.z * ql.z + ql.w * ql.w;
    float ssa = qa.x * qa.x + qa.y * qa.y + qa.z * qa.z + qa.w * qa.w;
    #pragma unroll
    for (int off = 16; off > 0; off >>= 1) {
        ssl += __shfl_xor(ssl, off, 32);
        ssa += __shfl_xor(ssa, off, 32);
    }
    const float rl = rsqrtf(ssl);
    const float ra = rsqrtf(ssa);

    *(v4f*)(new_l  + mo) = ql * rl;
    *(v4f*)(new_ab + mo) = qa * ra;
}

// ---------------------------------------------------------------------------
extern "C" void kernel_launch(void* const* d_in, const int* in_sizes, int n_in,
                              void* d_out, int out_size, void* d_ws, size_t ws_size,
                              hipStream_t stream)
{
    const float* l      = (const float*)d_in[0];
    const float* ab     = (const float*)d_in[1];
    const float* mem_l  = (const float*)d_in[2];
    const float* mem_ab = (const float*)d_in[3];
    const int*   y      = (const int*)d_in[4];
    const int*   idx    = (const int*)d_in[5];

    float* out    = (float*)d_out;
    float* out_l  = out;                                  // (B, K+1, 1)
    float* out_ab = out + (long)B_ * KP1;                 // (B, K+1, 1)
    float* new_l  = out + 2L * B_ * KP1;                  // (N, D)
    float* new_ab = new_l + (long)N_ * D_;                // (N, D)

    // 1025 tiles of 16 k-slots, 8 waves (tiles) per 256-thread block.
    dim3 sgrid((KP1 + 16 * 8 - 1) / (16 * 8), B_);
    nce_score_kernel<<<sgrid, 256, 0, stream>>>(l, ab, mem_l, mem_ab, idx,
                                                out_l, out_ab);

    bank_copy_kernel<<<2048, 256, 0, stream>>>((const v4f*)mem_l,
                                               (const v4f*)mem_ab,
                                               (v4f*)new_l, (v4f*)new_ab);

    ema_update_kernel<<<(B_ * 32 + 63) / 64, 64, 0, stream>>>(
        l, ab, mem_l, mem_ab, y, new_l, new_ab);
}